// Attention_67276367725070
// MI455X (gfx1250) — compile-verified
//
#include <hip/hip_runtime.h>
#include <hip/hip_bf16.h>

typedef __attribute__((ext_vector_type(2))) float v2f;
typedef __attribute__((ext_vector_type(8))) float v8f;

#define B_  32
#define S_  2048
#define E_  1024
#define D_  1024

// ---------------------------------------------------------------------------
// Small GEMM via V_WMMA_F32_16X16X4_F32:  O[32,1024] = H[32,1024] @ B(k,n)
// MODE 0: B(k,n) = W[k*1024 + n]   (u = h @ Wa_w), plain store
// MODE 1: B(k,n) = W[n*1024 + k]   (hp = h @ Wp_w^T), epilogue tanh(x + bias[n])
// One wave per 16x16 output tile: 2 M-tiles * 64 N-tiles = 128 waves = 16 blocks.
// ---------------------------------------------------------------------------
template <int MODE>
__global__ void small_gemm_wmma(const float* __restrict__ H,
                                const float* __restrict__ W,
                                const float* __restrict__ bias,
                                float* __restrict__ out) {
    const int lane = threadIdx.x & 31;
    const int wave = threadIdx.x >> 5;
    const int w    = blockIdx.x * 8 + wave;   // 0..127
    const int mtile = w & 1;                  // 0..1  (M = 32)
    const int ntile = w >> 1;                 // 0..63 (N = 1024)

    const int mrow  = mtile * 16 + (lane & 15);  // A row owned by this lane
    const int ncol  = ntile * 16 + (lane & 15);  // B col owned by this lane
    const int khalf = (lane >> 4) * 2;           // lanes 0-15: K=k0,k0+1; 16-31: k0+2,k0+3

    v8f acc = {};
    for (int k0 = 0; k0 < D_; k0 += 4) {
        const int ka = k0 + khalf;
        // A fragment: H[mrow, ka], H[mrow, ka+1] (contiguous)
        const float2 av = *(const float2*)(H + mrow * D_ + ka);
        v2f a; a[0] = av.x; a[1] = av.y;
        v2f b;
        if (MODE == 1) {
            const float2 bv = *(const float2*)(W + ncol * D_ + ka);
            b[0] = bv.x; b[1] = bv.y;
        } else {
            b[0] = W[ka * E_ + ncol];
            b[1] = W[(ka + 1) * E_ + ncol];
        }
        acc = __builtin_amdgcn_wmma_f32_16x16x4_f32(
            /*neg_a=*/false, a, /*neg_b=*/false, b,
            /*c_mod=*/(short)0, acc, /*reuse_a=*/false, /*reuse_b=*/false);
    }

    // C/D layout: VGPR r -> row (r + 8*(lane/16)), col = lane%16
    const int rbase = mtile * 16 + ((lane >> 4) << 3);
#pragma unroll
    for (int r = 0; r < 8; ++r) {
        float v = acc[r];
        if (MODE == 1) v = tanhf(v + bias[ncol]);
        out[(rbase + r) * E_ + ncol] = v;
    }
}

__device__ __forceinline__ float wave_sum(float v) {
#pragma unroll
    for (int off = 16; off > 0; off >>= 1) v += __shfl_xor(v, off, 32);
    return v;
}

// p[b] = S * sigmoid(hp[b,:]·vp_w + vp_b) ; c[b] = h[b,:]·Wa_b
__global__ void finish_pos(const float* __restrict__ hp, const float* __restrict__ h,
                           const float* __restrict__ vp_w, const float* __restrict__ vp_b,
                           const float* __restrict__ Wa_b,
                           float* __restrict__ p, float* __restrict__ c) {
    const int lane = threadIdx.x & 31;
    const int b = blockIdx.x * 8 + (threadIdx.x >> 5);   // grid 4 * 8 waves = 32
    float s1 = 0.f, s2 = 0.f;
    for (int d = lane; d < D_; d += 32) {
        s1 = fmaf(hp[b * D_ + d], vp_w[d], s1);
        s2 = fmaf(h[b * D_ + d], Wa_b[d], s2);
    }
    s1 = wave_sum(s1);
    s2 = wave_sum(s2);
    if (lane == 0) {
        const float z = s1 + vp_b[0];
        p[b] = (float)S_ / (1.0f + expf(-z));
        c[b] = s2;
    }
}

// att[b,s] = enc[b,s,:]·u[b,:] + c[b]   (streaming pass 1 over encoder_out)
__global__ void att_kernel(const float* __restrict__ enc, const float* __restrict__ u,
                           const float* __restrict__ c, float* __restrict__ att) {
    __shared__ float su[E_];
    const int tid  = threadIdx.x;
    const int lane = tid & 31;
    const int wave = tid >> 5;
    const int b    = blockIdx.y;

    // stage u[b,:] in LDS (4 KB)
    {
        const float4* usrc = (const float4*)(u + (size_t)b * E_);
        float4* udst = (float4*)su;
        udst[tid] = usrc[tid];   // 256 threads * float4 = 1024 floats
    }
    __syncthreads();
    const float cb = c[b];

    const int srow0 = blockIdx.x * 64 + wave * 8;
#pragma unroll
    for (int i = 0; i < 8; ++i) {
        const int srow = srow0 + i;
        const float* erow = enc + ((size_t)b * S_ + srow) * E_;
        float acc = 0.f;
#pragma unroll
        for (int j = 0; j < 8; ++j) {
            const int e = j * 128 + lane * 4;
            const float4 ev = *(const float4*)(erow + e);
            const float4 uv = *(const float4*)(su + e);
            acc = fmaf(ev.x, uv.x, acc);
            acc = fmaf(ev.y, uv.y, acc);
            acc = fmaf(ev.z, uv.z, acc);
            acc = fmaf(ev.w, uv.w, acc);
        }
        acc = wave_sum(acc);
        if (lane == 0) att[(size_t)b * S_ + srow] = acc + cb;
    }
}

// softmax over S then Gaussian window; one block per batch
__global__ void softmax_window(const float* __restrict__ att, const float* __restrict__ p,
                               float* __restrict__ alpha) {
    __shared__ float sh_e[S_];
    __shared__ float red[256];
    const int b = blockIdx.x;
    const int tid = threadIdx.x;
    const float pb = p[b];

    float m = -3.402823466e+38f;
    for (int s = tid; s < S_; s += 256) {
        const float v = att[(size_t)b * S_ + s];
        sh_e[s] = v;
        m = fmaxf(m, v);
    }
    red[tid] = m;
    __syncthreads();
#pragma unroll
    for (int o = 128; o > 0; o >>= 1) {
        if (tid < o) red[tid] = fmaxf(red[tid], red[tid + o]);
        __syncthreads();
    }
    const float mx = red[0];
    __syncthreads();

    float sum = 0.f;
    for (int s = tid; s < S_; s += 256) {
        const float e = expf(sh_e[s] - mx);
        sh_e[s] = e;
        sum += e;
    }
    red[tid] = sum;
    __syncthreads();
#pragma unroll
    for (int o = 128; o > 0; o >>= 1) {
        if (tid < o) red[tid] += red[tid + o];
        __syncthreads();
    }
    const float inv = 1.0f / red[0];

    for (int s = tid; s < S_; s += 256) {
        const float d = (float)s - pb;
        const float g = expf(-(d * d) * 0.125f);   // 1/(2*WINDOW_WIDTH^2), W=2
        alpha[(size_t)b * S_ + s] = sh_e[s] * inv * g;
    }
}

// awe partials: streaming pass 2 over encoder_out; S split 8-way (deterministic)
__global__ void awe_partial(const float* __restrict__ enc, const float* __restrict__ alpha,
                            float* __restrict__ partial) {
    __shared__ float sa[256];
    const int tid = threadIdx.x;
    const int echunk = blockIdx.x;   // 0..3
    const int schunk = blockIdx.y;   // 0..7
    const int b      = blockIdx.z;   // 0..31
    const int e  = echunk * 256 + tid;
    const int s0 = schunk * 256;

    sa[tid] = alpha[(size_t)b * S_ + s0 + tid];
    __syncthreads();

    float acc = 0.f;
    const float* base = enc + ((size_t)b * S_ + s0) * E_ + e;
#pragma unroll 4
    for (int i = 0; i < 256; ++i)
        acc = fmaf(base[(size_t)i * E_], sa[i], acc);

    partial[((size_t)(b * 8 + schunk)) * E_ + e] = acc;
}

__global__ void awe_reduce(const float* __restrict__ partial, float* __restrict__ awe) {
    const int idx = blockIdx.x * 256 + threadIdx.x;  // 0..32767
    const int b = idx >> 10;
    const int e = idx & 1023;
    float s = 0.f;
#pragma unroll
    for (int k = 0; k < 8; ++k) s += partial[((size_t)(b * 8 + k)) * E_ + e];
    awe[idx] = s;
}

extern "C" void kernel_launch(void* const* d_in, const int* in_sizes, int n_in,
                              void* d_out, int out_size, void* d_ws, size_t ws_size,
                              hipStream_t stream) {
    (void)in_sizes; (void)n_in; (void)out_size; (void)ws_size;

    const float* enc   = (const float*)d_in[0];  // [32,2048,1024]
    const float* h     = (const float*)d_in[1];  // [32,1024]
    /* d_in[2] = currentPosition : unused by reference */
    const float* Wa_w  = (const float*)d_in[3];  // [1024,1024]
    const float* Wa_b  = (const float*)d_in[4];  // [1024]
    const float* Wp_w  = (const float*)d_in[5];  // [1024,1024]
    const float* Wp_b  = (const float*)d_in[6];  // [1024]
    const float* vp_w  = (const float*)d_in[7];  // [1024]
    const float* vp_b  = (const float*)d_in[8];  // [1]

    float* out_awe   = (float*)d_out;            // [32,1024]
    float* out_alpha = out_awe + B_ * E_;        // [32,2048]

    float* ws = (float*)d_ws;
    float* ws_hp      = ws;  ws += B_ * D_;      // 32768
    float* ws_u       = ws;  ws += B_ * E_;      // 32768
    float* ws_c       = ws;  ws += B_;           // 32
    float* ws_p       = ws;  ws += B_;           // 32
    float* ws_att     = ws;  ws += B_ * S_;      // 65536
    float* ws_partial = ws;                      // 32*8*1024

    // 1. hp = tanh(h @ Wp_w^T + Wp_b)   [WMMA f32 16x16x4]
    small_gemm_wmma<1><<<16, 256, 0, stream>>>(h, Wp_w, Wp_b, ws_hp);
    // 2. u = h @ Wa_w                    [WMMA f32 16x16x4]
    small_gemm_wmma<0><<<16, 256, 0, stream>>>(h, Wa_w, nullptr, ws_u);
    // 3. p[b], c[b]
    finish_pos<<<4, 256, 0, stream>>>(ws_hp, h, vp_w, vp_b, Wa_b, ws_p, ws_c);
    // 4. att[b,s] = enc·u + c           (stream enc, pass 1)
    att_kernel<<<dim3(S_ / 64, B_), 256, 0, stream>>>(enc, ws_u, ws_c, ws_att);
    // 5. alpha = softmax(att) * gauss(p)
    softmax_window<<<B_, 256, 0, stream>>>(ws_att, ws_p, out_alpha);
    // 6-7. awe = enc^T·alpha            (stream enc, pass 2; deterministic split-K)
    awe_partial<<<dim3(E_ / 256, 8, B_), 256, 0, stream>>>(enc, out_alpha, ws_partial);
    awe_reduce<<<(B_ * E_) / 256, 256, 0, stream>>>(ws_partial, out_awe);
}